// MultiLevelRoIAligner_32719060861128
// MI455X (gfx1250) — compile-verified
//
#include <hip/hip_runtime.h>
#include <cstdint>

#define CROP 7
#define CC   256      // channels
#define NBOX 512      // boxes per batch
#define NPTS 28       // 2 rows x 14 x-taps staged per output row

typedef int v4i __attribute__((ext_vector_type(4)));   // 128-bit payload type

// ---- CDNA5 async global->LDS gather (ASYNCcnt path) ---------------------
__device__ __forceinline__ void async_copy_b128(const float* gptr, float* lptr) {
#if __has_builtin(__builtin_amdgcn_global_load_async_to_lds_b128)
  __builtin_amdgcn_global_load_async_to_lds_b128(
      (v4i*)gptr, (v4i*)lptr, /*offset=*/0, /*cpol=*/0);
#else
  // VDST = 32-bit LDS byte address, VADDR = 64-bit global address, SADDR=off.
  uint32_t loff = (uint32_t)(uint64_t)(uintptr_t)lptr;
  asm volatile("global_load_async_to_lds_b128 %0, %1, off"
               :: "v"(loff), "v"(gptr)
               : "memory");
#endif
}

template <int N>
__device__ __forceinline__ void wait_asynccnt() {
#if __has_builtin(__builtin_amdgcn_s_wait_asynccnt)
  __builtin_amdgcn_s_wait_asynccnt((unsigned short)N);
#else
  asm volatile("s_wait_asynccnt %0" :: "i"(N) : "memory");
#endif
}

// ---- kernel -------------------------------------------------------------
__global__ void __launch_bounds__(CC)
roialign_ml_kernel(const float* __restrict__ f2, const float* __restrict__ f3,
                   const float* __restrict__ f4, const float* __restrict__ f5,
                   const float* __restrict__ f6,
                   const float* __restrict__ boxes,
                   float* __restrict__ out) {
  // double-buffered gather tile: 2 x 28 points x 256 channels (57344 B)
  // 16-B aligned so B128 async writes land on architecturally aligned LDS.
  __shared__ __align__(16) float tile[2][NPTS][CC];
  __shared__ int xcol_s[14];

  const int bn = blockIdx.x;          // box id in [0, B*N)
  const int b  = bn >> 9;             // / 512
  const int t  = threadIdx.x;         // channel

  // ---- per-box scalars (uniform across the block; scalar-float pipe) ----
  const float4 box = ((const float4*)boxes)[bn];   // y1, x1, y2, x2
  const float bh = box.z - box.x;
  const float bw = box.w - box.y;
  const float asq = sqrtf(bh * bw);
  int lvl = (int)floorf(log2f(asq * (1.0f / 224.0f))) + 4;
  lvl = min(max(lvl, 2), 6);
  const float inv_scale = 1.0f / (float)(1 << lvl);
  const float by  = box.x * inv_scale;
  const float bx  = box.y * inv_scale;
  const float bhl = bh * inv_scale;
  const float bwl = bw * inv_scale;
  const int   li  = lvl - 2;
  const int   w   = 256 >> li;                 // level size == bound+1
  const float bound = (float)(w - 1);

  const float* feat;
  switch (li) {
    case 0:  feat = f2; break;
    case 1:  feat = f3; break;
    case 2:  feat = f4; break;
    case 3:  feat = f5; break;
    default: feat = f6; break;
  }
  const float* fbase = feat + (size_t)b * (size_t)w * (size_t)w * CC;

  // ---- x taps: weights in registers (static indexing), columns in LDS ----
  float wx[14];
#pragma unroll
  for (int j = 0; j < CROP; ++j) {
    const float gx = bx + (j + 0.5f) * bwl * (1.0f / 7.0f);
    const float x0 = fminf(fmaxf(floorf(gx), 0.0f), bound);
    const float lx = gx - x0;
    wx[2 * j]     = 1.0f - lx;
    wx[2 * j + 1] = lx;
  }
  if (t < CROP) {
    const float gx = bx + (t + 0.5f) * bwl * (1.0f / 7.0f);
    const float x0 = fminf(fmaxf(floorf(gx), 0.0f), bound);
    const float x1 = fminf(x0 + 1.0f, bound);
    xcol_s[2 * t]     = (int)x0;
    xcol_s[2 * t + 1] = (int)x1;
  }
  __syncthreads();

  // ---- async stage of one output row's 28 gather points (28 KB) ----
  auto stage = [&](int y, int buf) {
    const float gy  = by + (y + 0.5f) * bhl * (1.0f / 7.0f);
    const float y0f = fminf(fmaxf(floorf(gy), 0.0f), bound);
    const float y1f = fminf(y0f + 1.0f, bound);
    const int r0 = (int)y0f;
    const int r1 = (int)y1f;
#pragma unroll
    for (int k = 0; k < 7; ++k) {
      const int idx   = k * CC + t;        // 1792 lane-transfers of 16 B
      const int p     = idx >> 6;          // point 0..27 (64 transfers each)
      const int chunk = idx & 63;          // 16-B chunk within the 1-KB row
      const int ri    = (p >= 14) ? 1 : 0;
      const int xi    = ri ? (p - 14) : p;
      const int yy    = ri ? r1 : r0;
      const int col   = xcol_s[xi];
      const float* g  = fbase + ((size_t)yy * w + col) * CC + chunk * 4;
      float*       l  = &tile[buf][p][chunk * 4];
      async_copy_b128(g, l);
    }
  };

  float* outp = out + (size_t)bn * (CROP * CROP) * CC + t;

  stage(0, 0);
  for (int y = 0; y < CROP; ++y) {
    const int buf = y & 1;
    if (y < CROP - 1) stage(y + 1, buf ^ 1);       // prefetch next row
    if (y < CROP - 1) wait_asynccnt<7>();          // retire current row's 7 ops
    else              wait_asynccnt<0>();
    __syncthreads();                               // publish LDS across waves

    const float gy  = by + (y + 0.5f) * bhl * (1.0f / 7.0f);
    const float y0f = fminf(fmaxf(floorf(gy), 0.0f), bound);
    const float ly  = gy - y0f;
    const float ky0 = 1.0f - ly;
    const float ky1 = ly;

#pragma unroll
    for (int x = 0; x < CROP; ++x) {
      const float top = tile[buf][2 * x][t]      * wx[2 * x]
                      + tile[buf][2 * x + 1][t]  * wx[2 * x + 1];
      const float bot = tile[buf][14 + 2 * x][t]     * wx[2 * x]
                      + tile[buf][14 + 2 * x + 1][t] * wx[2 * x + 1];
      // NT store: don't let the 51 MB output stream evict the (L2-resident)
      // 178 MB feature pyramid on the 192 MB L2.
      __builtin_nontemporal_store(ky0 * top + ky1 * bot,
                                  &outp[(y * CROP + x) * CC]);
    }
    __syncthreads();   // all waves done reading before this buffer is restaged
  }
}

// ---- launcher -----------------------------------------------------------
extern "C" void kernel_launch(void* const* d_in, const int* in_sizes, int n_in,
                              void* d_out, int out_size, void* d_ws, size_t ws_size,
                              hipStream_t stream) {
  (void)in_sizes; (void)n_in; (void)out_size; (void)d_ws; (void)ws_size;
  const float* f2    = (const float*)d_in[0];
  const float* f3    = (const float*)d_in[1];
  const float* f4    = (const float*)d_in[2];
  const float* f5    = (const float*)d_in[3];
  const float* f6    = (const float*)d_in[4];
  const float* boxes = (const float*)d_in[5];
  float* out = (float*)d_out;

  const int n_blocks = 2 * NBOX;   // B * N = 1024 boxes, one block each
  roialign_ml_kernel<<<dim3(n_blocks), dim3(CC), 0, stream>>>(
      f2, f3, f4, f5, f6, boxes, out);
}